// Fast_WS_VectorQuantizer_12421045420170
// MI455X (gfx1250) — compile-verified
//
#include <hip/hip_runtime.h>
#include <hip/hip_bf16.h>
#include <math.h>

typedef __attribute__((ext_vector_type(2))) float v2f;
typedef __attribute__((ext_vector_type(8))) float v8f;

#define NTOT 4096      // N = B*H*W
#define DDIM 64
#define ZELEM 262144   // B*C*H*W = N*D
#define EPS 0.1f
#define BETA 0.25f

// ---------------------------------------------------------------------------
// Column means of w1/w2 (D x HK) and means of b1/b2.
// mean(x@W + b, axis=-1) == x . colmean(W) + mean(b)
// ---------------------------------------------------------------------------
__global__ __launch_bounds__(128) void prep_means(
    const float* __restrict__ w1, const float* __restrict__ b1,
    const float* __restrict__ w2, const float* __restrict__ b2,
    float* __restrict__ w1m, float* __restrict__ w2m, float* __restrict__ bm) {
  int t = threadIdx.x;
  if (t < 64) {
    float s = 0.f;
    for (int k = 0; k < 256; ++k) s += w1[t * 256 + k];
    w1m[t] = s * (1.0f / 256.0f);
  } else {
    int d = t - 64;
    float s = 0.f;
    for (int k = 0; k < 256; ++k) s += w2[d * 256 + k];
    w2m[d] = s * (1.0f / 256.0f);
  }
  if (t == 0) { float s = 0.f; for (int k = 0; k < 256; ++k) s += b1[k]; bm[0] = s * (1.0f/256.0f); }
  if (t == 1) { float s = 0.f; for (int k = 0; k < 256; ++k) s += b2[k]; bm[1] = s * (1.0f/256.0f); }
}

// ---------------------------------------------------------------------------
// Build zf (transposed z), zs = mu + exp(logvar)*noise, zero counts,
// and emit the z_q_noise output (which is exactly z).
// ---------------------------------------------------------------------------
__global__ __launch_bounds__(256) void build_inputs(
    const float* __restrict__ z, const float* __restrict__ codebook,
    const float* __restrict__ noise, float* __restrict__ zf,
    float* __restrict__ zs, int* __restrict__ counts, float* __restrict__ out) {
  int gid = blockIdx.x * 256 + threadIdx.x;     // 0 .. 262143
  int i = gid >> 6;                             // flat (h,w,b) index
  int d = gid & 63;
  int h = i >> 8, w = (i >> 4) & 15, b = i & 15;
  // z layout (B,C,H,W): z[((b*64 + d)*16 + h)*16 + w]
  zf[gid] = z[b * 16384 + d * 256 + h * 16 + w];
  // centroids = repeat(codebook, 8): row j -> codebook[j>>3]
  const float* cb = codebook + (i >> 3) * 128;
  zs[gid] = cb[d] + expf(cb[64 + d]) * noise[gid];
  // z_q_noise == z (double transpose is identity)
  out[ZELEM + gid] = z[gid];
  if (gid < NTOT) counts[gid] = 0;
}

// ---------------------------------------------------------------------------
// Per-row |x|^2 and phi = x . wmean + bmean. One wave per row.
// ---------------------------------------------------------------------------
__global__ __launch_bounds__(256) void row_stats(
    const float* __restrict__ src, const float* __restrict__ wm,
    const float* __restrict__ bm, float* __restrict__ sq, float* __restrict__ phi) {
  int lane = threadIdx.x & 31;
  int wave = threadIdx.x >> 5;
  int row = blockIdx.x * 8 + wave;
  float v0 = src[row * 64 + lane];
  float v1 = src[row * 64 + 32 + lane];
  float s = v0 * v0 + v1 * v1;
  float p = v0 * wm[lane] + v1 * wm[32 + lane];
  #pragma unroll
  for (int off = 16; off > 0; off >>= 1) {
    s += __shfl_xor(s, off, 32);
    p += __shfl_xor(p, off, 32);
  }
  if (lane == 0) { sq[row] = s; phi[row] = p + bm[0]; }
}

// ---------------------------------------------------------------------------
// Fused cost GEMM + row reductions via V_WMMA_F32_16X16X4_F32.
// Each wave owns 16 rows of A, streams all 4096 rows of B in 16-col tiles.
// cost[m][n] = sqA[m] + sqB[n] - 2 * (A_m . B_n)
// Keeps running argmin (if needIdx) and online logsumexp of
// (phiB[n] - cost)/EPS entirely in registers. EXEC stays all-ones.
// ---------------------------------------------------------------------------
__global__ __launch_bounds__(128) void cost_reduce(
    const float* __restrict__ A, const float* __restrict__ Bm,
    const float* __restrict__ sqA, const float* __restrict__ sqB,
    const float* __restrict__ phiB, float* __restrict__ lseOut,
    int* __restrict__ idxOut, int needIdx) {
  const int lane  = threadIdx.x & 31;
  const int wave  = threadIdx.x >> 5;
  const int i0    = (blockIdx.x * 4 + wave) * 16;
  const int mlane = lane & 15;
  const int khalf = (lane >> 4) * 2;   // k offset within 4-wide chunk
  const int mbase = (lane >> 4) * 8;   // accumulator row base for this half-wave

  // Preload the 16 A fragments (16x4 each) for this wave's row block.
  v2f a[16];
  const float* arow = A + (i0 + mlane) * 64 + khalf;
  #pragma unroll
  for (int kc = 0; kc < 16; ++kc) {
    a[kc].x = arow[kc * 4 + 0];
    a[kc].y = arow[kc * 4 + 1];
  }
  float sqa[8];
  #pragma unroll
  for (int v = 0; v < 8; ++v) sqa[v] = sqA[i0 + mbase + v];

  float minv[8], mx[8], ssum[8];
  int mini[8];
  #pragma unroll
  for (int v = 0; v < 8; ++v) {
    minv[v] = 3.4e38f; mini[v] = 0; mx[v] = -3.4e38f; ssum[v] = 0.f;
  }
  const float invEps = 1.0f / EPS;

  for (int j0 = 0; j0 < NTOT; j0 += 16) {
    const float* brow = Bm + (j0 + mlane) * 64 + khalf;
    v8f acc = {0.f, 0.f, 0.f, 0.f, 0.f, 0.f, 0.f, 0.f};
    #pragma unroll
    for (int kc = 0; kc < 16; ++kc) {
      v2f bfrag;
      bfrag.x = brow[kc * 4 + 0];
      bfrag.y = brow[kc * 4 + 1];
      acc = __builtin_amdgcn_wmma_f32_16x16x4_f32(
          false, a[kc], false, bfrag, (short)0, acc, false, false);
    }
    float sqb = sqB[j0 + mlane];
    float ph  = phiB[j0 + mlane];
    int colIdx = j0 + mlane;
    #pragma unroll
    for (int v = 0; v < 8; ++v) {
      float c = sqa[v] + sqb - 2.0f * acc[v];
      bool lt = c < minv[v];                 // v_cndmask, no EXEC change
      minv[v] = lt ? c : minv[v];
      mini[v] = lt ? colIdx : mini[v];
      float e = (ph - c) * invEps;
      float M = fmaxf(e, mx[v]);             // branchless online LSE
      ssum[v] = ssum[v] * __expf(mx[v] - M) + __expf(e - M);
      mx[v] = M;
    }
  }

  // Reduce across the 16 lanes that share each accumulator row.
  #pragma unroll
  for (int off = 1; off < 16; off <<= 1) {
    #pragma unroll
    for (int v = 0; v < 8; ++v) {
      float omin = __shfl_xor(minv[v], off, 32);
      int   oidx = __shfl_xor(mini[v], off, 32);
      bool take = (omin < minv[v]) || (omin == minv[v] && oidx < mini[v]);
      minv[v] = take ? omin : minv[v];
      mini[v] = take ? oidx : mini[v];
      float omx = __shfl_xor(mx[v], off, 32);
      float osm = __shfl_xor(ssum[v], off, 32);
      float M = fmaxf(mx[v], omx);
      ssum[v] = ssum[v] * __expf(mx[v] - M) + osm * __expf(omx - M);
      mx[v] = M;
    }
  }
  if (mlane == 0) {
    #pragma unroll
    for (int v = 0; v < 8; ++v) {
      int row = i0 + mbase + v;
      lseOut[row] = mx[v] + __logf(ssum[v]);
      if (needIdx) idxOut[row] = mini[v];
    }
  }
}

// ---------------------------------------------------------------------------
__global__ __launch_bounds__(256) void hist_kernel(
    const int* __restrict__ idx, int* __restrict__ counts) {
  int gid = blockIdx.x * 256 + threadIdx.x;
  if (gid < NTOT) atomicAdd(&counts[idx[gid]], 1);
}

// z_q output: gather mu[idx] and transpose back to (B,C,H,W).
__global__ __launch_bounds__(256) void zq_gather(
    const float* __restrict__ codebook, const int* __restrict__ idx,
    float* __restrict__ out) {
  int gid = blockIdx.x * 256 + threadIdx.x;   // 0..262143, (B,C,H,W) order
  int b = gid >> 14, c = (gid >> 8) & 63, h = (gid >> 4) & 15, w = gid & 15;
  int i = (h << 8) | (w << 4) | b;
  int j = idx[i];
  out[gid] = codebook[(j >> 3) * 128 + c];
}

// Scalars: loss and perplexity.
__global__ __launch_bounds__(256) void finalize(
    const float* __restrict__ lse1, const float* __restrict__ lse2,
    const float* __restrict__ phi1, const float* __restrict__ phi2,
    const int* __restrict__ counts, float* __restrict__ outs) {
  float s1 = 0, s2 = 0, p1 = 0, p2 = 0, ent = 0;
  for (int k = threadIdx.x; k < NTOT; k += 256) {
    s1 += lse1[k]; s2 += lse2[k]; p1 += phi1[k]; p2 += phi2[k];
    float e = (float)counts[k] * (1.0f / (float)NTOT);
    ent += e * logf(e + 1e-10f);
  }
  __shared__ float sm[5][256];
  sm[0][threadIdx.x] = s1; sm[1][threadIdx.x] = s2; sm[2][threadIdx.x] = p1;
  sm[3][threadIdx.x] = p2; sm[4][threadIdx.x] = ent;
  __syncthreads();
  for (int st = 128; st > 0; st >>= 1) {
    if (threadIdx.x < st)
      for (int q = 0; q < 5; ++q) sm[q][threadIdx.x] += sm[q][threadIdx.x + st];
    __syncthreads();
  }
  if (threadIdx.x == 0) {
    float n = (float)NTOT;
    float mp1 = sm[2][0] / n, mp2 = sm[3][0] / n;
    float logn = logf(n);
    // lse_ref1[i] = LSE1[i] - log n;  loss = (-EPS/n)*sum(LSE) + EPS*log n + mean(phi)
    float loss1 = (-EPS / n) * sm[0][0] + EPS * logn + mp1;
    float loss2 = (-EPS / n) * sm[1][0] + EPS * logn + mp2;
    outs[0] = BETA * (loss1 + loss2);
    outs[1] = expf(-sm[4][0]);
  }
}

// ---------------------------------------------------------------------------
extern "C" void kernel_launch(void* const* d_in, const int* in_sizes, int n_in,
                              void* d_out, int out_size, void* d_ws, size_t ws_size,
                              hipStream_t stream) {
  const float* z        = (const float*)d_in[0];
  const float* codebook = (const float*)d_in[1];
  // d_in[2] codebook_weight: unused by the reference computation
  const float* w1 = (const float*)d_in[3];
  const float* b1 = (const float*)d_in[4];
  const float* w2 = (const float*)d_in[5];
  const float* b2 = (const float*)d_in[6];
  const float* noise = (const float*)d_in[7];
  float* out = (float*)d_out;

  float* ws = (float*)d_ws;
  float* zf    = ws;                        // 262144
  float* zs    = ws + ZELEM;                // 262144
  float* sqzf  = ws + 2 * ZELEM;            // 4096
  float* sqzs  = sqzf + NTOT;               // 4096
  float* phi1  = sqzs + NTOT;               // 4096 (fn of zs)
  float* phi2  = phi1 + NTOT;               // 4096 (fn of zf)
  float* lse1  = phi2 + NTOT;               // 4096
  float* lse2  = lse1 + NTOT;               // 4096
  int*   counts = (int*)(lse2 + NTOT);      // 4096
  int*   idx    = counts + NTOT;            // 4096
  float* w1m   = (float*)(idx + NTOT);      // 64
  float* w2m   = w1m + 64;                  // 64
  float* bm    = w2m + 64;                  // 2 (b1mean, b2mean)

  prep_means<<<1, 128, 0, stream>>>(w1, b1, w2, b2, w1m, w2m, bm);
  build_inputs<<<ZELEM / 256, 256, 0, stream>>>(z, codebook, noise, zf, zs, counts, out);
  row_stats<<<NTOT / 8, 256, 0, stream>>>(zf, w2m, bm + 1, sqzf, phi2);
  row_stats<<<NTOT / 8, 256, 0, stream>>>(zs, w1m, bm + 0, sqzs, phi1);
  // Row direction: argmin + LSE over j of (phi1[j]-cost[i,j])/EPS
  cost_reduce<<<NTOT / 64, 128, 0, stream>>>(zf, zs, sqzf, sqzs, phi1, lse1, idx, 1);
  // Column direction (cost transposed): LSE over i of (phi2[i]-cost[i,j])/EPS
  cost_reduce<<<NTOT / 64, 128, 0, stream>>>(zs, zf, sqzs, sqzf, phi2, lse2, idx, 0);
  hist_kernel<<<NTOT / 256, 256, 0, stream>>>(idx, counts);
  zq_gather<<<ZELEM / 256, 256, 0, stream>>>(codebook, idx, out);
  finalize<<<1, 256, 0, stream>>>(lse1, lse2, phi1, phi2, counts, out + 2 * ZELEM);
}